// UniformQuantizer_42202348650890
// MI455X (gfx1250) — compile-verified
//
#include <hip/hip_runtime.h>
#include <stdint.h>
#include <stddef.h>

// MX-style group quantize/dequantize, GROUP=128, wave32-per-group.
// Memory-bound: ~512MB traffic -> ~22us floor @ 23.3 TB/s.
// CDNA5 paths used: global_load_async_to_lds_b128 (ASYNCcnt-tracked DMA into
// LDS, double buffered per wave), s_wait_asynccnt split counter, wave32
// shuffle-butterfly reduction, non-temporal b128 stores.

typedef float f32x4 __attribute__((ext_vector_type(4)));

#define GROUP_SIZE 128
#define LANES      32
#define WPB        8            // waves per block
#define TPB        (WPB * LANES)

// One async DMA: each of the 32 lanes copies 16B  global -> LDS (512B = 1 group).
__device__ __forceinline__ void async_group_load(uint32_t lds_off, uint32_t byte_off,
                                                 const float* base) {
    asm volatile("global_load_async_to_lds_b128 %0, %1, %2"
                 :
                 : "v"(lds_off), "v"(byte_off), "s"(base)
                 : "memory");
}

__global__ void __launch_bounds__(TPB, 1)
mx_group_quant_kernel(const float* __restrict__ x, float* __restrict__ out, int ngroups) {
    __shared__ f32x4 smem[WPB][2][LANES];   // 8KB: double buffer per wave

    const int lane    = threadIdx.x & (LANES - 1);
    const int wib     = threadIdx.x >> 5;                 // wave in block
    const int wave    = blockIdx.x * WPB + wib;           // global wave id
    const int wstride = gridDim.x * WPB;

    f32x4* o4 = (f32x4*)out;

    // Raw LDS byte offsets of this wave's two staging slots (low 32 bits of
    // the generic pointer are the LDS offset on gfx1250).
    const uint32_t lds0 = (uint32_t)(uintptr_t)&smem[wib][0][lane];
    const uint32_t lds1 = (uint32_t)(uintptr_t)&smem[wib][1][lane];

    int g = wave;
    if (g >= ngroups) return;

    // Prime the pipeline: group g -> buffer 0.
    async_group_load(lds0, (uint32_t)g * 512u + (uint32_t)lane * 16u, x);

    int buf = 0;
    for (; g < ngroups; g += wstride) {
        const int gn = g + wstride;
        if (gn < ngroups) {
            // Prefetch next group into the other buffer, then wait until only
            // that prefetch is outstanding (ASYNCcnt <= 1 => current group done;
            // async loads complete in issue order).
            async_group_load(buf ? lds0 : lds1,
                             (uint32_t)gn * 512u + (uint32_t)lane * 16u, x);
            asm volatile("s_wait_asynccnt 0x1" ::: "memory");
        } else {
            asm volatile("s_wait_asynccnt 0x0" ::: "memory");
        }

        const f32x4 v = smem[wib][buf][lane];

        // Per-lane absmax of 4 elements, then wave32 butterfly max (group absmax).
        float m = fmaxf(fmaxf(__builtin_fabsf(v.x), __builtin_fabsf(v.y)),
                        fmaxf(__builtin_fabsf(v.z), __builtin_fabsf(v.w)));
#pragma unroll
        for (int off = 16; off > 0; off >>= 1)
            m = fmaxf(m, __shfl_xor(m, off, 32));

        // shared_exp = clip(ceil(log2(m)) - 7, -128, 127), exact via bit tricks:
        // ceil(log2(m)) = unbiased_exp + (mantissa != 0) for normal m; all
        // zero/denormal cases fall below -128 and clamp identically.
        const uint32_t u = __float_as_uint(m);
        int e = (int)((u >> 23) & 0xFF) - 127 + ((u & 0x7FFFFFu) ? 1 : 0) - 7;
        e = e < -128 ? -128 : (e > 127 ? 127 : e);

        const float sc  = __builtin_ldexpf(1.0f, -e);   // 2^-shared_exp
        const float dsc = __builtin_ldexpf(1.0f,  e);   // 2^+shared_exp

        f32x4 r;
        r.x = fminf(127.0f, fmaxf(-128.0f, __builtin_rintf(v.x * sc))) * dsc;
        r.y = fminf(127.0f, fmaxf(-128.0f, __builtin_rintf(v.y * sc))) * dsc;
        r.z = fminf(127.0f, fmaxf(-128.0f, __builtin_rintf(v.z * sc))) * dsc;
        r.w = fminf(127.0f, fmaxf(-128.0f, __builtin_rintf(v.w * sc))) * dsc;

        // Write-once output: non-temporal b128 store, keep L2 for the read stream.
        __builtin_nontemporal_store(r, &o4[(size_t)g * LANES + lane]);

        buf ^= 1;
    }
}

extern "C" void kernel_launch(void* const* d_in, const int* in_sizes, int n_in,
                              void* d_out, int out_size, void* d_ws, size_t ws_size,
                              hipStream_t stream) {
    (void)n_in; (void)out_size; (void)d_ws; (void)ws_size;
    const float* x = (const float*)d_in[0];
    float* out = (float*)d_out;

    const long long n = in_sizes[0];
    const int ngroups = (int)(n / GROUP_SIZE);
    if (ngroups <= 0) return;

    int blocks = (ngroups + WPB - 1) / WPB;
    if (blocks > 4096) blocks = 4096;   // ~32K waves, 2 async loads in flight each

    mx_group_quant_kernel<<<dim3(blocks), dim3(TPB), 0, stream>>>(x, out, ngroups);
}